// HydraScaleLM_8572754723189
// MI455X (gfx1250) — compile-verified
//
#include <hip/hip_runtime.h>
#include <hip/hip_bf16.h>
#include <math.h>

// ---------------------------------------------------------------------------
// Model constants (match reference)
// ---------------------------------------------------------------------------
#define BB 2
#define LL 512
#define DD 1024
#define NST 16
#define KCONV 4
#define RR 64
#define DEPTH 4
#define VV 50000
#define MLPD 4096
#define BL (BB * LL)          // 1024 rows for all big GEMMs
#define PCOLS (RR + 2 * NST)  // 96

typedef __attribute__((ext_vector_type(2))) float v2f;
typedef __attribute__((ext_vector_type(8))) float v8f;

// ---------------------------------------------------------------------------
// WMMA fp32 GEMM:  C[M,N] = act(A[M,K] * W[N,K]^T + bias) (+ add_src)
//
// Memory-bound on the weight stream (M = 1024 only), so the tile is skewed
// toward M to minimize weight re-reads:
//   block = 256 threads = 8 wave32 waves, arranged 4 (M) x 2 (N)
//   wave tile  = 64 (M) x 32 (N)  -> 4x2 = 8 independent v8f accumulators
//   block tile = 256 (M) x 64 (N) -> weights streamed M/256 = 4 times total
//   K staged through LDS in chunks of 32 (sA 256x33, sW 64x33; ~42 KB)
// A (activations, <=16MB) re-reads come from L2 (192MB).
//
// The compute loop carries NO bounds checks: WMMA raises no exceptions, so
// out-of-range tiles compute on uninitialized LDS and are simply not stored.
// This keeps EXEC all-ones (WMMA requirement) and lets the compiler pipeline
// ds_load_2addr against back-to-back v_wmma with partial dscnt waits.
// act: 0=none 1=silu 2=gelu(exact) 3=softplus
// Requires: M % 256 == 0, K % 32 == 0, N % 16 == 0 (all true here).
// ---------------------------------------------------------------------------
#define BKC 32
#define MT  256
#define NT  64

__global__ __launch_bounds__(256) void gemm_wmma_f32(
    const float* __restrict__ A, const float* __restrict__ W,
    const float* __restrict__ bias, const float* __restrict__ add_src,
    float* __restrict__ C, int M, int N, int K, int lda, int ldc, int act)
{
    __shared__ float sA[MT][BKC + 1];
    __shared__ float sW[NT][BKC + 1];

    const int tid     = threadIdx.x;
    const int wave    = tid >> 5;
    const int lane    = tid & 31;
    const int l16     = lane & 15;
    const int halfsel = lane >> 4;             // 0: lanes 0-15, 1: lanes 16-31
    const int wm      = wave & 3;              // wave's M slot (0..3)
    const int wn      = wave >> 2;             // wave's N slot (0..1)
    const int rowbase = blockIdx.y * MT;
    const int colbase = blockIdx.x * NT;

    v8f acc[4][2] = {};

    for (int k0 = 0; k0 < K; k0 += BKC) {
        // ---- stage A tile: 256 x 32 floats, 8 float4 per thread -----------
        #pragma unroll
        for (int it = 0; it < 8; ++it) {
            int chunk = tid + it * 256;        // 0..2047 float4 chunks
            int r = chunk >> 3;                // 0..255
            int c = (chunk & 7) << 2;          // 0..28 step 4
            const float4 v =
                *(const float4*)(A + (size_t)(rowbase + r) * lda + k0 + c);
            sA[r][c + 0] = v.x; sA[r][c + 1] = v.y;
            sA[r][c + 2] = v.z; sA[r][c + 3] = v.w;
        }
        // ---- stage W tile: 64 x 32 floats, 2 float4 per thread ------------
        #pragma unroll
        for (int it = 0; it < 2; ++it) {
            int chunk = tid + it * 256;        // 0..511
            int r = chunk >> 3;                // 0..63
            int c = (chunk & 7) << 2;
            int gn = colbase + r;
            if (gn < N) {                      // no OOB reads; LDS rows for
                const float* src = W + (size_t)gn * K + k0 + c;  // gn>=N stay
                const float4 v = *(const float4*)src;            // undefined
                sW[r][c + 0] = v.x; sW[r][c + 1] = v.y;
                sW[r][c + 2] = v.z; sW[r][c + 3] = v.w;
                if (k0 + BKC < K)              // stream next weight chunk
                    __builtin_prefetch(src + BKC, 0, 1);
            }
        }
        __syncthreads();

        // ---- compute: 8 k-steps, 8 unconditional WMMAs per step -----------
        #pragma unroll
        for (int kk = 0; kk < BKC; kk += 4) {
            const int ko = kk + halfsel * 2;
            v2f b[2];
            #pragma unroll
            for (int nt = 0; nt < 2; ++nt) {
                const int nrow = wn * 32 + nt * 16 + l16;
                b[nt].x = sW[nrow][ko];
                b[nt].y = sW[nrow][ko + 1];
            }
            #pragma unroll
            for (int mt = 0; mt < 4; ++mt) {
                const int mrow = wm * 64 + mt * 16 + l16;
                v2f a;
                a.x = sA[mrow][ko];
                a.y = sA[mrow][ko + 1];
                #pragma unroll
                for (int nt = 0; nt < 2; ++nt) {
                    acc[mt][nt] = __builtin_amdgcn_wmma_f32_16x16x4_f32(
                        false, a, false, b[nt], (short)0, acc[mt][nt],
                        false, false);
                }
            }
        }
        __syncthreads();
    }

    // ---- epilogue (only place with N bounds checks) ------------------------
    #pragma unroll
    for (int nt = 0; nt < 2; ++nt) {
        const int ncol = colbase + wn * 32 + nt * 16;     // tile base
        if (ncol >= N) continue;       // tiles are fully valid or invalid
        const int n = ncol + l16;
        const float bv = bias ? bias[n] : 0.0f;
        #pragma unroll
        for (int mt = 0; mt < 4; ++mt) {
            #pragma unroll
            for (int v = 0; v < 8; ++v) {
                const int m = rowbase + wm * 64 + mt * 16 + v + halfsel * 8;
                float x = acc[mt][nt][v] + bv;
                if (act == 1) {                            // silu
                    x = x / (1.0f + __expf(-x));
                } else if (act == 2) {                     // exact gelu
                    x = 0.5f * x * (1.0f + erff(x * 0.70710678118654752f));
                } else if (act == 3) {                     // softplus
                    x = (x > 20.0f) ? x : log1pf(__expf(x));
                }
                if (add_src) x += add_src[(size_t)m * ldc + n];
                C[(size_t)m * ldc + n] = x;
            }
        }
    }
}

// ---------------------------------------------------------------------------
// LayerNorm: one 256-thread block per row of length DD
// ---------------------------------------------------------------------------
__global__ __launch_bounds__(256) void ln_kernel(
    const float* __restrict__ x, const float* __restrict__ sc,
    const float* __restrict__ bi, float* __restrict__ out, int Dd)
{
    const int row = blockIdx.x;
    const float* xr = x + (size_t)row * Dd;
    float sum = 0.0f, sq = 0.0f;
    for (int j = threadIdx.x; j < Dd; j += 256) {
        float v = xr[j]; sum += v; sq += v * v;
    }
    #pragma unroll
    for (int off = 16; off > 0; off >>= 1) {
        sum += __shfl_down(sum, off, 32);
        sq  += __shfl_down(sq,  off, 32);
    }
    __shared__ float rs[8], rq[8];
    __shared__ float s_mean, s_rstd;
    if ((threadIdx.x & 31) == 0) { rs[threadIdx.x >> 5] = sum; rq[threadIdx.x >> 5] = sq; }
    __syncthreads();
    if (threadIdx.x == 0) {
        float ts = 0.0f, tq = 0.0f;
        for (int i = 0; i < 8; ++i) { ts += rs[i]; tq += rq[i]; }
        float mean = ts / (float)Dd;
        float var  = tq / (float)Dd - mean * mean;
        s_mean = mean;
        s_rstd = rsqrtf(var + 1e-5f);
    }
    __syncthreads();
    const float mean = s_mean, rstd = s_rstd;
    float* orow = out + (size_t)row * Dd;
    for (int j = threadIdx.x; j < Dd; j += 256)
        orow[j] = (xr[j] - mean) * rstd * sc[j] + bi[j];
}

// ---------------------------------------------------------------------------
// Time embedding sin/cos table: e[b][k], k<512 -> sin, else cos
// ---------------------------------------------------------------------------
__global__ void sincos_kernel(const int* __restrict__ ts, float* __restrict__ e,
                              int half, int Dd)
{
    int i = blockIdx.x * 256 + threadIdx.x;
    if (i >= BB * Dd) return;
    int b = i / Dd, k = i % Dd;
    float t = (float)ts[b];
    const float c = -logf(10000.0f) / (float)(half - 1);
    float v;
    if (k < half) v = sinf(t * __expf((float)k * c));
    else          v = cosf(t * __expf((float)(k - half) * c));
    e[i] = v;
}

// Naive dense (for the tiny B=2 time-embedding MLP)
__global__ void dense_kernel(const float* __restrict__ A, const float* __restrict__ W,
                             const float* __restrict__ bias, float* __restrict__ out,
                             int Mr, int Nc, int Kin, int act)
{
    int i = blockIdx.x * 256 + threadIdx.x;
    if (i >= Mr * Nc) return;
    int mm = i / Nc, nn = i % Nc;
    const float* a = A + (size_t)mm * Kin;
    const float* w = W + (size_t)nn * Kin;
    float s = bias ? bias[nn] : 0.0f;
    for (int k = 0; k < Kin; ++k) s += a[k] * w[k];
    if (act == 1) s = s / (1.0f + __expf(-s));
    out[i] = s;
}

// x[b,l,:] = emb_table[ids[b,l]] + temb[b,:]
__global__ void embed_kernel(const int* __restrict__ ids,
                             const float* __restrict__ table,
                             const float* __restrict__ temb,
                             float* __restrict__ x, int Lc, int Dd)
{
    size_t i = (size_t)blockIdx.x * 256 + threadIdx.x;
    int d = (int)(i % Dd);
    size_t bl = i / Dd;
    int b = (int)(bl / Lc);
    int id = ids[bl];
    x[i] = table[(size_t)id * Dd + d] + temb[(size_t)b * Dd + d];
}

// Depthwise causal conv (K=4) + SiLU: u[b,l,d]
__global__ void conv_silu_kernel(const float* __restrict__ xres,
                                 const float* __restrict__ cw,
                                 float* __restrict__ u, int Lc, int Dd)
{
    size_t i = (size_t)blockIdx.x * 256 + threadIdx.x;
    int d = (int)(i % Dd);
    size_t bl = i / Dd;
    int l = (int)(bl % Lc);
    int b = (int)(bl / Lc);
    float s = 0.0f;
    #pragma unroll
    for (int j = 0; j < KCONV; ++j) {
        int ll = l - (KCONV - 1) + j;
        if (ll >= 0)
            s += cw[d * KCONV + j] * xres[((size_t)b * Lc + ll) * Dd + d];
    }
    u[i] = s / (1.0f + __expf(-s));
}

// ---------------------------------------------------------------------------
// Selective scan: one lane per (b,d); h[N=16] in registers; b/c projections
// broadcast per timestep through LDS. Writes y = (scan + u*Dp)*silu(z).
// grid: (DD/256, BB), block 256
// ---------------------------------------------------------------------------
__global__ __launch_bounds__(256) void scan_kernel(
    const float* __restrict__ dt, const float* __restrict__ p,
    const float* __restrict__ u, const float* __restrict__ z,
    const float* __restrict__ a_log, const float* __restrict__ Dp,
    float* __restrict__ y, int Lc, int Dd, int ldp)
{
    const int b = blockIdx.y;
    const int d = blockIdx.x * 256 + threadIdx.x;
    __shared__ float sB[NST], sC[NST];

    float a[NST], h[NST];
    #pragma unroll
    for (int n = 0; n < NST; ++n) {
        a[n] = -__expf(a_log[(size_t)d * NST + n]);
        h[n] = 0.0f;
    }
    const float dp = Dp[d];

    for (int l = 0; l < Lc; ++l) {
        const float* prow = p + ((size_t)b * Lc + l) * ldp;
        if (threadIdx.x < NST)              sB[threadIdx.x] = prow[RR + threadIdx.x];
        else if (threadIdx.x < 2 * NST)     sC[threadIdx.x - NST] = prow[RR + threadIdx.x];
        __syncthreads();

        const size_t idx = ((size_t)b * Lc + l) * Dd + d;
        const float dtv = dt[idx], uv = u[idx], zv = z[idx];
        float yv = 0.0f;
        #pragma unroll
        for (int n = 0; n < NST; ++n) {
            const float at = __expf(dtv * a[n]);
            const float bt = (fabsf(a[n]) < 1e-8f) ? dtv
                             : (at - 1.0f) / (a[n] + 1e-10f);
            h[n] = at * h[n] + bt * sB[n] * uv;
            yv  += sC[n] * h[n];
        }
        yv = (yv + uv * dp) * (zv / (1.0f + __expf(-zv)));
        y[idx] = yv;
        __syncthreads();
    }
}

// ---------------------------------------------------------------------------
// Host-side launch helpers
// ---------------------------------------------------------------------------
static inline void gemm(hipStream_t s, const float* A, const float* W,
                        const float* bias, const float* add, float* C,
                        int M, int N, int K, int lda, int ldc, int act)
{
    dim3 grid((N + NT - 1) / NT, M / MT);
    gemm_wmma_f32<<<grid, 256, 0, s>>>(A, W, bias, add, C, M, N, K, lda, ldc, act);
}

extern "C" void kernel_launch(void* const* d_in, const int* in_sizes, int n_in,
                              void* d_out, int out_size, void* d_ws, size_t ws_size,
                              hipStream_t stream)
{
    const int*   input_ids = (const int*)d_in[0];
    const int*   timesteps = (const int*)d_in[1];
    const float* emb_table = (const float*)d_in[2];
    const float* tw1       = (const float*)d_in[3];
    const float* tb1       = (const float*)d_in[4];
    const float* tw2       = (const float*)d_in[5];
    const float* tb2       = (const float*)d_in[6];
    const float* ln1_s     = (const float*)d_in[7];
    const float* ln1_b     = (const float*)d_in[8];
    const float* wx        = (const float*)d_in[9];
    const float* wz        = (const float*)d_in[10];
    const float* wp        = (const float*)d_in[11];
    const float* conv_w    = (const float*)d_in[12];
    const float* dt_w      = (const float*)d_in[13];
    const float* dt_b      = (const float*)d_in[14];
    const float* A_log     = (const float*)d_in[15];
    const float* Dp        = (const float*)d_in[16];
    const float* wout      = (const float*)d_in[17];
    const float* ln2_s     = (const float*)d_in[18];
    const float* ln2_b     = (const float*)d_in[19];
    const float* mlp_w1    = (const float*)d_in[20];
    const float* mlp_b1    = (const float*)d_in[21];
    const float* mlp_w2    = (const float*)d_in[22];
    const float* mlp_b2    = (const float*)d_in[23];
    const float* lnout_s   = (const float*)d_in[24];
    const float* lnout_b   = (const float*)d_in[25];
    const float* head_w    = (const float*)d_in[26];
    const float* head_b    = (const float*)d_in[27];
    float* logits = (float*)d_out;

    // ---- workspace carve-up (floats) ----
    float* ws = (float*)d_ws;
    size_t off = 0;
    const size_t BLD = (size_t)BL * DD;
    float* xbuf  = ws + off; off += BLD;            // residual stream x
    float* xn    = ws + off; off += BLD;            // LN output
    float* xres  = ws + off; off += BLD;
    float* zbuf  = ws + off; off += BLD;
    float* ubuf  = ws + off; off += BLD;
    float* ybuf  = ws + off; off += BLD;
    float* dtbuf = ws + off; off += BLD;
    float* pbuf  = ws + off; off += (size_t)BL * PCOLS;
    float* mh    = ws + off; off += (size_t)BL * MLPD;
    float* e0    = ws + off; off += (size_t)BB * DD;
    float* t1    = ws + off; off += (size_t)BB * 4 * DD;
    float* temb  = ws + off; off += (size_t)BB * DD;

    const int HALF = DD / 2;
    const size_t total = BLD;
    const int eblocks = (int)((total + 255) / 256);

    // ---- time embedding (tiny, B=2) ----
    sincos_kernel<<<(BB * DD + 255) / 256, 256, 0, stream>>>(timesteps, e0, HALF, DD);
    dense_kernel<<<(BB * 4 * DD + 255) / 256, 256, 0, stream>>>(
        e0, tw1, tb1, t1, BB, 4 * DD, DD, /*silu*/1);
    dense_kernel<<<(BB * DD + 255) / 256, 256, 0, stream>>>(
        t1, tw2, tb2, temb, BB, DD, 4 * DD, /*none*/0);

    // ---- embedding gather + broadcast add ----
    embed_kernel<<<eblocks, 256, 0, stream>>>(input_ids, emb_table, temb, xbuf, LL, DD);

    // ---- layers ----
    for (int i = 0; i < DEPTH; ++i) {
        const float* wx_i  = wx     + (size_t)i * DD * DD;
        const float* wz_i  = wz     + (size_t)i * DD * DD;
        const float* wp_i  = wp     + (size_t)i * PCOLS * DD;
        const float* cw_i  = conv_w + (size_t)i * DD * KCONV;
        const float* dtw_i = dt_w   + (size_t)i * DD * RR;
        const float* dtb_i = dt_b   + (size_t)i * DD;
        const float* al_i  = A_log  + (size_t)i * DD * NST;
        const float* dp_i  = Dp     + (size_t)i * DD;
        const float* wo_i  = wout   + (size_t)i * DD * DD;
        const float* w1_i  = mlp_w1 + (size_t)i * MLPD * DD;
        const float* b1_i  = mlp_b1 + (size_t)i * MLPD;
        const float* w2_i  = mlp_w2 + (size_t)i * DD * MLPD;
        const float* b2_i  = mlp_b2 + (size_t)i * DD;

        // LN1
        ln_kernel<<<BL, 256, 0, stream>>>(xbuf, ln1_s + (size_t)i * DD,
                                          ln1_b + (size_t)i * DD, xn, DD);
        // projections
        gemm(stream, xn, wx_i, nullptr, nullptr, xres, BL, DD, DD, DD, DD, 0);
        gemm(stream, xn, wz_i, nullptr, nullptr, zbuf, BL, DD, DD, DD, DD, 0);
        gemm(stream, xn, wp_i, nullptr, nullptr, pbuf, BL, PCOLS, DD, DD, PCOLS, 0);
        // dt = softplus(dt_u @ dtw^T + dtb) ; dt_u = pbuf[:, :R] (lda = 96)
        gemm(stream, pbuf, dtw_i, dtb_i, nullptr, dtbuf, BL, DD, RR, PCOLS, DD, 3);
        // depthwise conv + silu
        conv_silu_kernel<<<eblocks, 256, 0, stream>>>(xres, cw_i, ubuf, LL, DD);
        // selective scan -> ybuf (includes u*Dp skip and silu(z) gate)
        scan_kernel<<<dim3(DD / 256, BB), 256, 0, stream>>>(
            dtbuf, pbuf, ubuf, zbuf, al_i, dp_i, ybuf, LL, DD, PCOLS);
        // out-proj + residual add: x = x + y @ wout^T
        gemm(stream, ybuf, wo_i, nullptr, xbuf, xbuf, BL, DD, DD, DD, DD, 0);
        // LN2 + MLP
        ln_kernel<<<BL, 256, 0, stream>>>(xbuf, ln2_s + (size_t)i * DD,
                                          ln2_b + (size_t)i * DD, xn, DD);
        gemm(stream, xn, w1_i, b1_i, nullptr, mh, BL, MLPD, DD, DD, MLPD, /*gelu*/2);
        gemm(stream, mh, w2_i, b2_i, xbuf, xbuf, BL, DD, MLPD, MLPD, DD, 0);
    }

    // ---- final LN + vocab head (memory-bound: streams 205MB of head_w) ----
    ln_kernel<<<BL, 256, 0, stream>>>(xbuf, lnout_s, lnout_b, xn, DD);
    gemm(stream, xn, head_w, head_b, nullptr, logits, BL, VV, DD, DD, VV, 0);
}